// SurvEMD_90941637525552
// MI455X (gfx1250) — compile-verified
//
#include <hip/hip_runtime.h>
#include <stdint.h>

// SurvEMD forward, B=1048576 rows x 64 bins -> scalar mean loss.
// Streaming, memory-bound (~264 MB @ 23.3 TB/s ~= 11.5us floor).
// CDNA5 path: async global->LDS staging (ASYNCcnt, triple-buffered,
// distance-2 prefetch) + DPP16/readlane wave32 scan + SMEM t/e loads.

#define NB 64
#define TOTAL_ROWS 1048576
#define ROWS_PER_WAVE 64
#define STAGE_ROWS 8
#define NSTAGES 8                         // 64 rows / 8 per stage
#define WAVES_PER_BLOCK 8
#define BLOCK_THREADS 256
#define NUM_BLOCKS 2048                   // 2048*8 waves * 64 rows = 1048576
#define STAGE_FLOATS (STAGE_ROWS * NB)    // 512 floats = 2048 B
#define STAGE_BYTES  (STAGE_FLOATS * 4)
#define NBUF 3                            // triple buffer
#define WAVE_LDS_FLOATS (NBUF * STAGE_FLOATS)
#define WAVE_LDS_BYTES  (NBUF * STAGE_BYTES)   // 6 KB per wave

typedef __attribute__((address_space(3))) const char lds_cchar_t;

// Raw LDS byte offset (what the async-load VDST field wants).
__device__ __forceinline__ unsigned lds_offset_of(const void* p) {
  return (unsigned)(unsigned long long)(lds_cchar_t*)p;
}

// Copy one 2KB stage (8 rows of 64 floats) global -> LDS asynchronously.
// 4 x B128: each lane moves 16B; imm offset applies to both LDS and global.
__device__ __forceinline__ void async_stage(const float* gbase,
                                            unsigned lds_byte,
                                            unsigned lane) {
  unsigned ldsa = lds_byte + lane * 16u;
  unsigned voff = lane * 16u;
  unsigned long long sb = (unsigned long long)(uintptr_t)gbase;
  asm volatile(
      "global_load_async_to_lds_b128 %0, %1, %2 offset:0\n\t"
      "global_load_async_to_lds_b128 %0, %1, %2 offset:512\n\t"
      "global_load_async_to_lds_b128 %0, %1, %2 offset:1024\n\t"
      "global_load_async_to_lds_b128 %0, %1, %2 offset:1536"
      :: "v"(ldsa), "v"(voff), "s"(sb)
      : "memory");
}

// x shifted right by N lanes within each 16-lane row, zero fill (DPP16).
template <int N>
__device__ __forceinline__ float dpp_row_shr(float v) {
  return __int_as_float(__builtin_amdgcn_update_dpp(
      0, __float_as_int(v), 0x110 + N, 0xF, 0xF, true));
}

__device__ __forceinline__ float readlane_f(float v, int lane) {
  return __int_as_float(__builtin_amdgcn_readlane(__float_as_int(v), lane));
}

__global__ __launch_bounds__(BLOCK_THREADS)
void survemd_main(const float* __restrict__ y,
                  const int* __restrict__ t,
                  const int* __restrict__ e,
                  float* __restrict__ part) {
  __shared__ __align__(16) char smem_raw[WAVES_PER_BLOCK * WAVE_LDS_BYTES + 64];
  float* smem = (float*)smem_raw;
  float* redbuf = (float*)(smem_raw + WAVES_PER_BLOCK * WAVE_LDS_BYTES);

  const unsigned lane = threadIdx.x & 31u;
  const unsigned wib  = (unsigned)__builtin_amdgcn_readfirstlane((int)(threadIdx.x >> 5));
  const unsigned gwave = blockIdx.x * WAVES_PER_BLOCK + wib;
  const size_t base_row = (size_t)gwave * ROWS_PER_WAVE;   // uniform
  const float* grow = y + base_row * NB;

  float* wave_lds = smem + (size_t)wib * WAVE_LDS_FLOATS;
  const unsigned wave_lds_byte = lds_offset_of(wave_lds);

  const int j0 = 2 * (int)lane;
  const int j1 = j0 + 1;
  const float eA   = 22026.465794806718f;     // exp(+10)
  const float eB   = 4.5399929762484854e-05f; // exp(-10)
  const float eAmB = eA - eB;
  const float c0 = (float)(j0 + 1) * eB;
  const float c1 = (float)(j1 + 1) * eB;
  const float zB = 64.0f * eB;

  float acc = 0.0f;

  // Process one landed 8-row stage out of LDS. t/e addresses are uniform
  // (base_row from readfirstlane/blockIdx, sbase a loop scalar) -> the
  // compiler emits clustered SMEM s_load's onto the scalar pipe.
  auto process_stage = [&](const float* buf, int sbase) {
    int tts[STAGE_ROWS], ets[STAGE_ROWS];
    #pragma unroll
    for (int r2 = 0; r2 < STAGE_ROWS; ++r2) {
      tts[r2] = t[base_row + (size_t)(sbase + r2)];
      ets[r2] = e[base_row + (size_t)(sbase + r2)];
    }
    #pragma unroll
    for (int r2 = 0; r2 < STAGE_ROWS; ++r2) {
      const int tt = tts[r2];
      const bool cens = (ets[r2] != 0);  // e==1: censored -> pred = y_hat

      const float2 yv = *(const float2*)(buf + r2 * NB + j0);
      const float p0 = (!cens && j0 >= tt) ? 10.0f : yv.x;
      const float p1 = (!cens && j1 >= tt) ? 10.0f : yv.y;

      const float w0 = __expf(p0);
      const float w1 = __expf(p1);
      const float sl = w0 + w1;

      // Inclusive prefix sum across the wave: DPP16 scan + half patch.
      float x = sl;
      x += dpp_row_shr<1>(x);
      x += dpp_row_shr<2>(x);
      x += dpp_row_shr<4>(x);
      x += dpp_row_shr<8>(x);                 // per-16-lane inclusive scan
      const float low15 = readlane_f(x, 15);  // sum of lanes 0..15
      x += (lane >= 16u) ? low15 : 0.0f;      // inclusive over all 32
      const float S = readlane_f(x, 31);

      const float excl = x - sl;
      const float invS = __builtin_amdgcn_rcpf(S);
      const float P0   = (excl + w0) * invS;  // cumsum(pred_dist) at j0
      const float P1   = x * invS;            // at j1

      // Analytic cumsum(target_dist): ones form a suffix (e=0) or {t} (e=1).
      const int   kk   = cens ? 1 : (NB - tt);
      const float Z    = fmaf((float)kk, eAmB, zB);
      const float invZ = __builtin_amdgcn_rcpf(Z);
      const int o0 = cens ? (int)(j0 >= tt) : max(0, j0 - tt + 1);
      const int o1 = cens ? (int)(j1 >= tt) : max(0, j1 - tt + 1);
      const float T0 = fmaf((float)o0, eAmB, c0) * invZ;
      const float T1 = fmaf((float)o1, eAmB, c1) * invZ;

      const float d0 = P0 - T0;
      const float d1 = P1 - T1;
      acc = fmaf(d0, d0, acc);
      acc = fmaf(d1, d1, acc);
    }
  };

  // Distance-2 prefetch pipeline (stages 0 and 1 in flight).
  async_stage(grow,                wave_lds_byte,               lane);
  async_stage(grow + STAGE_FLOATS, wave_lds_byte + STAGE_BYTES, lane);

  // Steady state: single unconditional issue + wait per iteration.
  int bpre = 2;  // buffer receiving stage s+2
  int bcur = 0;  // buffer holding stage s
  for (int s = 0; s < NSTAGES - 2; ++s) {
    async_stage(grow + (size_t)(s + 2) * STAGE_FLOATS,
                wave_lds_byte + (unsigned)bpre * STAGE_BYTES, lane);
    // 3 stages (12 ops) possibly in flight; <=8 left => stage s landed.
    asm volatile("s_wait_asynccnt 8" ::: "memory");
    process_stage(wave_lds + bcur * STAGE_FLOATS, s * STAGE_ROWS);
    bpre = (bpre == NBUF - 1) ? 0 : bpre + 1;
    bcur = (bcur == NBUF - 1) ? 0 : bcur + 1;
  }
  // Epilogue: stages 6 (buffer 0) and 7 (buffer 1).
  asm volatile("s_wait_asynccnt 4" ::: "memory");
  process_stage(wave_lds + bcur * STAGE_FLOATS, (NSTAGES - 2) * STAGE_ROWS);
  bcur = (bcur == NBUF - 1) ? 0 : bcur + 1;
  asm volatile("s_wait_asynccnt 0" ::: "memory");
  process_stage(wave_lds + bcur * STAGE_FLOATS, (NSTAGES - 1) * STAGE_ROWS);

  // Wave tree-reduce, then block reduce into per-block partial.
  #pragma unroll
  for (int d = 16; d > 0; d >>= 1) acc += __shfl_xor(acc, d, 32);
  if (lane == 0) redbuf[wib] = acc;
  __syncthreads();
  if (threadIdx.x == 0) {
    float b = 0.0f;
    #pragma unroll
    for (int i = 0; i < WAVES_PER_BLOCK; ++i) b += redbuf[i];
    part[blockIdx.x] = b;
  }
}

__global__ __launch_bounds__(BLOCK_THREADS)
void survemd_reduce(const float* __restrict__ part, float* __restrict__ out) {
  __shared__ float sred[WAVES_PER_BLOCK];
  float a = 0.0f;
  for (int i = threadIdx.x; i < NUM_BLOCKS; i += BLOCK_THREADS) a += part[i];
  #pragma unroll
  for (int d = 16; d > 0; d >>= 1) a += __shfl_xor(a, d, 32);
  const unsigned lane = threadIdx.x & 31u;
  const unsigned wib  = threadIdx.x >> 5;
  if (lane == 0) sred[wib] = a;
  __syncthreads();
  if (threadIdx.x == 0) {
    float tot = 0.0f;
    #pragma unroll
    for (int i = 0; i < WAVES_PER_BLOCK; ++i) tot += sred[i];
    out[0] = tot * (1.0f / (float)TOTAL_ROWS);
  }
}

extern "C" void kernel_launch(void* const* d_in, const int* in_sizes, int n_in,
                              void* d_out, int out_size, void* d_ws, size_t ws_size,
                              hipStream_t stream) {
  const float* y = (const float*)d_in[0];
  const int*   t = (const int*)d_in[1];
  const int*   e = (const int*)d_in[2];
  float* part = (float*)d_ws;  // NUM_BLOCKS floats of scratch

  survemd_main<<<NUM_BLOCKS, BLOCK_THREADS, 0, stream>>>(y, t, e, part);
  survemd_reduce<<<1, BLOCK_THREADS, 0, stream>>>(part, (float*)d_out);
}